// SymbolicFeatureExtractor_88648124990559
// MI455X (gfx1250) — compile-verified
//
#include <hip/hip_runtime.h>
#include <hip/hip_bf16.h>

#define MAXLEN 128
#define HASHSZ 256
#define EMPTYK 0xFFFFFFFFu

typedef __attribute__((ext_vector_type(2))) float v2f;
typedef __attribute__((ext_vector_type(8))) float v8f;

// ---------------------------------------------------------------------------
// Kernel 1: per-row features.  One wave32 per row, 4 tokens per lane.
// Distinct-count via a per-row 256-entry LDS hash set with linear probing
// (ds_cmpstore_rtn_b32).  O(L) work per row instead of O(L^2) sort/compare,
// so we stay near the 23.3 TB/s bandwidth roofline (~12.5 us for 290 MB).
// ---------------------------------------------------------------------------
__global__ __launch_bounds__(256) void feats_kernel(
    const int* __restrict__ ids, const int* __restrict__ am,
    float* __restrict__ feats, int fstride, int B)
{
    __shared__ unsigned tab[8][HASHSZ];
    const int wave = threadIdx.x >> 5;
    const int lane = threadIdx.x & 31;
    const int row  = blockIdx.x * 8 + wave;

    unsigned* t = tab[wave];
#pragma unroll
    for (int i = 0; i < HASHSZ / 32; ++i) t[lane + 32 * i] = EMPTYK;
    __syncthreads();

    if (row < B) {
        const int4 v = ((const int4*)(ids + (size_t)row * MAXLEN))[lane];
        const int4 m = ((const int4*)(am  + (size_t)row * MAXLEN))[lane];
        const int vals[4] = { v.x, v.y, v.z, v.w };
        const int msk[4]  = { m.x, m.y, m.z, m.w };
        int nvalid = 0, nnew = 0;
#pragma unroll
        for (int k = 0; k < 4; ++k) {
            if (msk[k] != 0) {
                ++nvalid;
                const unsigned key = (unsigned)vals[k];
                unsigned h = (key * 2654435761u) >> 24;   // 8-bit hash
                for (;;) {
                    const unsigned old = atomicCAS(&t[h], EMPTYK, key);
                    if (old == EMPTYK) { ++nnew; break; } // first time this value seen
                    if (old == key)    { break; }         // duplicate value
                    h = (h + 1) & (HASHSZ - 1);           // probe (128 keys < 256 slots)
                }
            }
        }
        // wave32 butterfly reduction
#pragma unroll
        for (int off = 16; off > 0; off >>= 1) {
            nvalid += __shfl_xor(nvalid, off, 32);
            nnew   += __shfl_xor(nnew,   off, 32);
        }
        if (lane == 0) {
            const float s = (float)nvalid, u = (float)nnew;
            const float ratio = (nvalid > 0) ? (u / s) : 0.0f;
            const float4 f = make_float4(s * (1.0f / 128.0f), u * (1.0f / 128.0f), ratio, 0.0f);
            *(float4*)(feats + (size_t)row * fstride) = f;
        }
    }
}

// ---------------------------------------------------------------------------
// Kernel 2: MLP 3->32->32 with ReLU, fp32 WMMA.  One wave32 per 16-row tile.
// Layer 1: A = feats 16x4 (K=3 zero padded), B = W1 4x16 tiles -> 2 WMMAs.
// D-layout -> A-layout transpose through LDS (stride 34 to avoid conflicts).
// Layer 2: K=32 -> 8 chained V_WMMA_F32_16X16X4_F32 per 16-col tile, x2 tiles.
// ---------------------------------------------------------------------------
__global__ __launch_bounds__(256) void mlp_wmma_kernel(
    const float* __restrict__ feats, int fstride,
    const float* __restrict__ W1, const float* __restrict__ b1v,
    const float* __restrict__ W2, const float* __restrict__ b2v,
    float* __restrict__ out, int B)
{
    __shared__ float sh[8][16 * 34];
    const int wave   = threadIdx.x >> 5;
    const int lane   = threadIdx.x & 31;
    const int r0     = (blockIdx.x * 8 + wave) * 16;
    const int lo     = lane & 15;
    const bool hiH   = lane >= 16;          // lanes 16..31 carry K=2,3 (A) / rows M+8 (C/D)
    const bool active = (r0 + 16) <= B;
    const int mbase  = hiH ? 8 : 0;
    float* S = sh[wave];

    if (active) {
        // A1: 16x4 f32.  lane<16: (f0,f1) of row r0+lane; lane>=16: (f2, 0).
        const float4 f = *(const float4*)(feats + (size_t)(r0 + lo) * fstride);
        v2f a1;
        a1.x = hiH ? f.z : f.x;
        a1.y = hiH ? 0.0f : f.y;
#pragma unroll
        for (int t = 0; t < 2; ++t) {
            const int col = t * 16 + lo;
            // B1: 4x16 f32.  lane<16: rows K0,K1 of W1; lane>=16: K2, zero-pad K3.
            v2f bm;
            bm.x = W1[(hiH ? 2 : 0) * 32 + col];
            bm.y = hiH ? 0.0f : W1[32 + col];
            v8f c = {};
            v8f d = __builtin_amdgcn_wmma_f32_16x16x4_f32(
                false, a1, false, bm, (short)0, c, false, false);
            const float bias = b1v[col];
#pragma unroll
            for (int r = 0; r < 8; ++r) {
                float x = d[r] + bias;
                x = x > 0.0f ? x : 0.0f;
                S[(mbase + r) * 34 + col] = x;   // h[m][n], row stride 34
            }
        }
    }
    __syncthreads();
    if (active) {
        // Re-load h in A-layout: lane<16 needs h[lo][4kc+0..1], lane>=16 h[lo][4kc+2..3]
        v2f a2[8];
#pragma unroll
        for (int kc = 0; kc < 8; ++kc) {
            const int base = lo * 34 + kc * 4 + (hiH ? 2 : 0);
            a2[kc].x = S[base];
            a2[kc].y = S[base + 1];
        }
#pragma unroll
        for (int t = 0; t < 2; ++t) {
            const int col = t * 16 + lo;
            v8f acc = {};
#pragma unroll
            for (int kc = 0; kc < 8; ++kc) {
                const int krow = kc * 4 + (hiH ? 2 : 0);
                v2f bm;
                bm.x = W2[krow * 32 + col];
                bm.y = W2[(krow + 1) * 32 + col];
                acc = __builtin_amdgcn_wmma_f32_16x16x4_f32(
                    false, a2[kc], false, bm, (short)0, acc, false, false);
            }
            const float bias = b2v[col];
#pragma unroll
            for (int r = 0; r < 8; ++r) {
                float x = acc[r] + bias;
                x = x > 0.0f ? x : 0.0f;
                out[(size_t)(r0 + mbase + r) * 32 + col] = x;
            }
        }
    }
}

extern "C" void kernel_launch(void* const* d_in, const int* in_sizes, int n_in,
                              void* d_out, int out_size, void* d_ws, size_t ws_size,
                              hipStream_t stream)
{
    const int*   ids = (const int*)d_in[0];
    const int*   am  = (const int*)d_in[1];
    const float* W1  = (const float*)d_in[2];
    const float* b1  = (const float*)d_in[3];
    const float* W2  = (const float*)d_in[4];
    const float* b2  = (const float*)d_in[5];
    float* out = (float*)d_out;
    const int B = in_sizes[0] / MAXLEN;

    // Stage features in d_ws (float4 per row).  If ws is too small, stage them
    // inside d_out at stride 32: each wave in kernel 2 reads its own 16 rows'
    // features before overwriting those same rows, so this is race-free.
    float* feats;
    int fstride;
    if (ws_size >= (size_t)B * 4 * sizeof(float)) {
        feats = (float*)d_ws; fstride = 4;
    } else {
        feats = out; fstride = 32;
    }

    const dim3 blk(256);
    const dim3 g1((unsigned)((B + 7) / 8));       // 8 rows (waves) per block
    feats_kernel<<<g1, blk, 0, stream>>>(ids, am, feats, fstride, B);

    const dim3 g2((unsigned)((B + 127) / 128));   // 8 waves x 16 rows per block
    mlp_wmma_kernel<<<g2, blk, 0, stream>>>(feats, fstride, W1, b1, W2, b2, out, B);
}